// Triplet_163208757671
// MI455X (gfx1250) — compile-verified
//
#include <hip/hip_runtime.h>

// Triplet loss, algebraically reduced:
//   S_pos   = sum_c || sum_{i in class c} z_i ||^2   (WMMA f32 Gram diagonal)
//   S_total = || sum_i z_i ||^2
//   out = max(S_pos/n_pos - (S_total-S_pos)/n_neg + alpha, 0)

#define B_ROWS   8192
#define D_DIM    256
#define N_CLS    512
#define ALPHA_C  0.1f

typedef __attribute__((ext_vector_type(2))) float v2f;
typedef __attribute__((ext_vector_type(8))) float v8f;

// ---------------------------------------------------------------- init ----
__global__ void init_ws_kernel(int* __restrict__ ws, int nwords) {
    int i = blockIdx.x * blockDim.x + threadIdx.x;
    if (i < nwords) ws[i] = 0;   // zero bits == 0.0f and int 0
}

// ------------------------------------------------- normalize + scatter ----
// One wave32 per row: lane j holds elements j, j+32, ..., j+224.
__global__ void class_accum_kernel(const int* __restrict__ y_true,
                                   const float* __restrict__ y_pred,
                                   float* __restrict__ cls,
                                   int* __restrict__ counts) {
    const int wave = threadIdx.x >> 5;          // 0..7
    const int lane = threadIdx.x & 31;
    const int row  = blockIdx.x * 8 + wave;     // grid = B_ROWS/8 blocks

    const float* xr = y_pred + (size_t)row * D_DIM;
    float x[8];
    float ss = 0.f;
#pragma unroll
    for (int j = 0; j < 8; ++j) {
        x[j] = xr[lane + 32 * j];
        ss += x[j] * x[j];
    }
#pragma unroll
    for (int off = 16; off; off >>= 1) ss += __shfl_xor(ss, off);

    // divide_no_nan semantics: zero row -> zero output
    const float inv = (ss > 0.f) ? (1.f / sqrtf(ss)) : 0.f;

    const int lab = y_true[row];
    float* cr = cls + (size_t)lab * D_DIM;
#pragma unroll
    for (int j = 0; j < 8; ++j)
        unsafeAtomicAdd(&cr[lane + 32 * j], x[j] * inv);   // global_atomic_add_f32

    if (lane == 0) atomicAdd(&counts[lab], 1);
}

// ------------------------------------- WMMA f32 Gram-diagonal (S_pos) ----
// One wave per 16-class tile: C(16x16) = A(16x256) * A(16x256)^T via 64
// V_WMMA_F32_16X16X4_F32 steps.  For f32 operand layouts the register image
// of A and of B=A^T coincide, so the same v2f feeds both operands.
__global__ void gram_diag_wmma_kernel(const float* __restrict__ cls,
                                      float* __restrict__ s_pos_accum) {
    const int lane = threadIdx.x;       // blockDim.x == 32, EXEC all ones
    const int l    = lane & 15;         // M for A, N for B/C
    const int h    = lane >> 4;         // lane-half selects K offset (A/B), M+8 (C)
    const int c0   = blockIdx.x * 16;   // grid = N_CLS/16 = 32 blocks

    const float* rowp = cls + (size_t)(c0 + l) * D_DIM + 2 * h;

    v8f c = {};                          // f32 accumulator, 8 VGPRs
    for (int kk = 0; kk < D_DIM; kk += 4) {
        v2f a;
        a.x = rowp[kk];                  // K = kk + 2h
        a.y = rowp[kk + 1];              // K = kk + 2h + 1
        c = __builtin_amdgcn_wmma_f32_16x16x4_f32(
                /*neg_a=*/false, a, /*neg_b=*/false, a,
                /*c_mod=*/(short)0, c, /*reuse_a=*/false, /*reuse_b=*/false);
    }

    // Diagonal of C: lane L holds C[r+8h][l] in c[r].  Diagonal lives in
    // lanes 0..7 (c[l]) and lanes 24..31 (c[l-8]).
    int want = -1;
    if (h == 0 && l < 8)  want = l;
    if (h == 1 && l >= 8) want = l - 8;
    float diag = 0.f;
#pragma unroll
    for (int r = 0; r < 8; ++r)
        if (want == r) diag = c[r];

#pragma unroll
    for (int off = 16; off; off >>= 1) diag += __shfl_xor(diag, off);
    if (lane == 0) unsafeAtomicAdd(s_pos_accum, diag);
}

// ------------------------------------------------------------ finalize ----
__global__ void finalize_kernel(const float* __restrict__ cls,
                                const int* __restrict__ counts,
                                const float* __restrict__ s_pos_accum,
                                float* __restrict__ out) {
    __shared__ float red_st[8];
    __shared__ float red_np[8];
    const int tid = threadIdx.x;        // 256 threads

    // t[d] = sum_c cls[c][d]  (coalesced: contiguous across threads)
    float t = 0.f;
    for (int cc = 0; cc < N_CLS; ++cc) t += cls[(size_t)cc * D_DIM + tid];
    float st = t * t;                   // partial of S_total = ||t||^2

    const float c0 = (float)counts[tid];
    const float c1 = (float)counts[tid + 256];
    float np = c0 * c0 + c1 * c1;       // partial of n_pos

#pragma unroll
    for (int off = 16; off; off >>= 1) {
        st += __shfl_xor(st, off);
        np += __shfl_xor(np, off);
    }
    if ((tid & 31) == 0) { red_st[tid >> 5] = st; red_np[tid >> 5] = np; }
    __syncthreads();

    if (tid == 0) {
        float S_total = 0.f, n_pos = 0.f;
#pragma unroll
        for (int w = 0; w < 8; ++w) { S_total += red_st[w]; n_pos += red_np[w]; }
        const float S_pos = s_pos_accum[0];
        const float S_neg = S_total - S_pos;
        const float n_neg = (float)B_ROWS * (float)B_ROWS - n_pos;
        const float res   = S_pos / n_pos - S_neg / n_neg + ALPHA_C;
        out[0] = res > 0.f ? res : 0.f;
    }
}

// -------------------------------------------------------------- launch ----
extern "C" void kernel_launch(void* const* d_in, const int* in_sizes, int n_in,
                              void* d_out, int out_size, void* d_ws, size_t ws_size,
                              hipStream_t stream) {
    const int*   y_true = (const int*)d_in[0];     // [B,1] int32
    const float* y_pred = (const float*)d_in[1];   // [B,D] f32

    // Workspace layout: cls_sum[512*256] f32 | counts[512] i32 | accum[8] f32
    float* cls    = (float*)d_ws;
    int*   counts = (int*)(cls + N_CLS * D_DIM);
    float* accum  = (float*)(counts + N_CLS);
    const int nwords = N_CLS * D_DIM + N_CLS + 8;   // ~514 KB

    init_ws_kernel<<<(nwords + 255) / 256, 256, 0, stream>>>((int*)d_ws, nwords);
    class_accum_kernel<<<B_ROWS / 8, 256, 0, stream>>>(y_true, y_pred, cls, counts);
    gram_diag_wmma_kernel<<<N_CLS / 16, 32, 0, stream>>>(cls, accum);
    finalize_kernel<<<1, 256, 0, stream>>>(cls, counts, accum, (float*)d_out);
}